// OptimizedVelocityField_56590489092611
// MI455X (gfx1250) — compile-verified
//
#include <hip/hip_runtime.h>
#include <math.h>

// ---------------------------------------------------------------------------
// OptimizedVelocityField for MI455X (gfx1250, wave32).
//
// Algebraic reduction: softmax logits are affine in pairwise distance, so the
// whole O(P*HID) attention collapses to {min,max,Σexp,Σd·exp} reductions over
// P = N(N-1)/2 distances plus a rank-1 update. The MLP (1.8M MACs) is done by
// a single 512-thread workgroup using fp32 WMMA (V_WMMA_F32_16X16X4_F32),
// which preserves fp32 reference numerics while using the matrix pipe.
// ---------------------------------------------------------------------------

namespace {

constexpr int   kN     = 1024;
constexpr int   kHID   = 256;
constexpr int   kHeads = 4;
constexpr int   kHD    = 64;
constexpr int   kDim   = 3072;
constexpr int   kMlpIn = 3137;   // kDim + kHD + 1
constexpr float kEps   = 1e-5f;

// ws layout (floats): [0]=dmin bits, [1]=dmax bits, [2..5]=a_h,
//                     [6..9]=sum_exp[h], [10..13]=sum_dexp[h]
typedef float v2f __attribute__((ext_vector_type(2)));
typedef float v8f __attribute__((ext_vector_type(8)));

__device__ __forceinline__ float gelu_tanh(float v) {
  // jax.nn.gelu default (approximate=True)
  return 0.5f * v * (1.0f + tanhf(0.7978845608028654f * (v + 0.044715f * v * v * v)));
}

// One 16-output tile of y = W(rows i0..i0+15) · vin, contraction length kc
// (multiple of 4), via V_WMMA_F32_16X16X4_F32.
// A 16x4 layout (ISA 7.12.2): lane L -> row M = L&15; k-half = (L>>4)*2, so
// VGPR0/1 hold K={kb,kb+1}. B 4x16 mirrors the k mapping; since B is the
// broadcast vector vin, the N striping is irrelevant.
__device__ __forceinline__ v8f wmma_matvec16(const float* __restrict__ W, int ld,
                                             int i0, const float* __restrict__ vin,
                                             int kc) {
  const int lane = threadIdx.x & 31;
  const int m    = lane & 15;
  const int kb   = (lane >> 4) << 1;          // 0 or 2
  const float* rowp = W + (size_t)(i0 + m) * ld + kb;
  v8f c = {};
  for (int j0 = 0; j0 < kc; j0 += 4) {
    v2f a, b;
    a.x = rowp[j0];
    a.y = rowp[j0 + 1];
    b.x = vin[j0 + kb];
    b.y = vin[j0 + kb + 1];
    c = __builtin_amdgcn_wmma_f32_16x16x4_f32(false, a, false, b, (short)0, c,
                                              false, false);
  }
  return c;
}

// --------------------------- kernel 1: init --------------------------------
__global__ void k_init(const float* __restrict__ Wk, const float* __restrict__ t,
                       const float* __restrict__ Wq, const float* __restrict__ bq,
                       float* __restrict__ wsf, unsigned* __restrict__ wsu) {
  __shared__ float sm[256];
  const int tid = threadIdx.x;                 // 256 threads
  const float q = Wq[tid] * t[0] + bq[tid];
  sm[tid] = Wk[tid] * q;                       // per-feature product
  __syncthreads();
  // segmented reduce: 4 heads of 64 contiguous features each
  for (int s = 32; s > 0; s >>= 1) {
    if ((tid & 63) < s) sm[tid] += sm[tid + s];
    __syncthreads();
  }
  if ((tid & 63) == 0) wsf[2 + (tid >> 6)] = sm[tid];   // a_h
  if (tid == 0) { wsu[0] = 0x7f800000u; wsu[1] = 0u; }  // +inf / 0 (bit order ok, d>=0)
  if (tid < 8) wsf[6 + tid] = 0.0f;                     // zero sums
}

// ----------------- kernel 2: distance min/max (recompute) ------------------
// Block b covers rows {b, 1022-b}: each block gets exactly ~1024 pairs.
__global__ void k_dist(const float* __restrict__ x, unsigned* __restrict__ wsu) {
  __shared__ float smin[256], smax[256];
  const int tid = threadIdx.x;
  float lmin = 3.0e38f, lmax = 0.0f;
  int rows[2] = { (int)blockIdx.x, 1022 - (int)blockIdx.x };
  const int nr = (rows[0] == rows[1]) ? 1 : 2;
  for (int r = 0; r < nr; ++r) {
    const int i = rows[r];
    const float xi0 = x[3 * i], xi1 = x[3 * i + 1], xi2 = x[3 * i + 2];
    for (int j = i + 1 + tid; j < kN; j += 256) {
      const float d0 = xi0 - x[3 * j];
      const float d1 = xi1 - x[3 * j + 1];
      const float d2 = xi2 - x[3 * j + 2];
      const float d = sqrtf(d0 * d0 + d1 * d1 + d2 * d2);
      lmin = fminf(lmin, d);
      lmax = fmaxf(lmax, d);
    }
  }
  smin[tid] = lmin; smax[tid] = lmax;
  __syncthreads();
  for (int s = 128; s > 0; s >>= 1) {
    if (tid < s) {
      smin[tid] = fminf(smin[tid], smin[tid + s]);
      smax[tid] = fmaxf(smax[tid], smax[tid + s]);
    }
    __syncthreads();
  }
  if (tid == 0) {
    atomicMin(&wsu[0], __float_as_uint(smin[0]));
    atomicMax(&wsu[1], __float_as_uint(smax[0]));
  }
}

// --------------- kernel 3: per-head exp-sum reductions ---------------------
__global__ void k_exps(const float* __restrict__ x, float* __restrict__ wsf) {
  __shared__ float sred[256];
  const int tid = threadIdx.x;
  const float dmin = wsf[0], dmax = wsf[1];
  float ah[kHeads], mh[kHeads];
#pragma unroll
  for (int h = 0; h < kHeads; ++h) {
    ah[h] = wsf[2 + h];
    mh[h] = (ah[h] >= 0.0f) ? ah[h] * dmax : ah[h] * dmin;  // max logit shift
  }
  float se[kHeads] = {0, 0, 0, 0}, sde[kHeads] = {0, 0, 0, 0};
  int rows[2] = { (int)blockIdx.x, 1022 - (int)blockIdx.x };
  const int nr = (rows[0] == rows[1]) ? 1 : 2;
  for (int r = 0; r < nr; ++r) {
    const int i = rows[r];
    const float xi0 = x[3 * i], xi1 = x[3 * i + 1], xi2 = x[3 * i + 2];
    for (int j = i + 1 + tid; j < kN; j += 256) {
      const float d0 = xi0 - x[3 * j];
      const float d1 = xi1 - x[3 * j + 1];
      const float d2 = xi2 - x[3 * j + 2];
      const float d = sqrtf(d0 * d0 + d1 * d1 + d2 * d2);
#pragma unroll
      for (int h = 0; h < kHeads; ++h) {
        const float e = __expf(ah[h] * d - mh[h]);   // <= 1, no overflow
        se[h] += e;
        sde[h] += d * e;
      }
    }
  }
  float vals[8] = { se[0], se[1], se[2], se[3], sde[0], sde[1], sde[2], sde[3] };
  for (int q = 0; q < 8; ++q) {
    sred[tid] = vals[q];
    __syncthreads();
    for (int s = 128; s > 0; s >>= 1) {
      if (tid < s) sred[tid] += sred[tid + s];
      __syncthreads();
    }
    if (tid == 0) atomicAdd(&wsf[6 + q], sred[0]);
    __syncthreads();
  }
}

// ------------- kernel 4: fused pair_feat + MLP (single WG, WMMA) -----------
__device__ __forceinline__ void rmsnorm_gelu(const float* __restrict__ sY,
                                             float* __restrict__ sH,
                                             const float* __restrict__ gw,
                                             float* __restrict__ sRed, int tid) {
  const float v = (tid < kHID) ? sY[tid] : 0.0f;
  sRed[tid] = v * v;
  __syncthreads();
  for (int s = 256; s > 0; s >>= 1) {
    if (tid < s) sRed[tid] += sRed[tid + s];
    __syncthreads();
  }
  const float rs = rsqrtf(sRed[0] * (1.0f / (float)kHID) + kEps);
  if (tid < kHID) sH[tid] = gelu_tanh(v * rs * gw[tid]);
  __syncthreads();
}

__global__ __launch_bounds__(512)
void k_mlp(const float* __restrict__ x, const float* __restrict__ t,
           const float* __restrict__ Wv, const float* __restrict__ bv,
           const float* __restrict__ W0, const float* __restrict__ b0,
           const float* __restrict__ Wh, const float* __restrict__ bh,
           const float* __restrict__ Wout, const float* __restrict__ bout,
           const float* __restrict__ g, const float* __restrict__ ws,
           float* __restrict__ out) {
  __shared__ float sFeat[kMlpIn];
  __shared__ float sY[kHID];
  __shared__ float sH[kHID];
  __shared__ float sRed[512];
  const int tid  = threadIdx.x;
  const int wave = tid >> 5;           // 16 waves
  const int lane = tid & 31;
  const float tval = t[0];

  // pair_feat = S * Wv[:,0] + H * bv   (softmax weights sum to 1)
  float S = 0.0f;
#pragma unroll
  for (int h = 0; h < kHeads; ++h) S += ws[10 + h] / ws[6 + h];

  for (int i = tid; i < kDim; i += 512) sFeat[i] = x[i];
  if (tid < kHD) sFeat[kDim + tid] = S * Wv[tid] + (float)kHeads * bv[tid];
  if (tid == 0) sFeat[kDim + kHD] = tval;

  // Warm L2 with Wout (needed last) while layer 0 runs: global_prefetch_b8.
  {
    const char* p = (const char*)Wout;
    const size_t bytes = (size_t)kDim * kHID * sizeof(float);
    for (size_t off = (size_t)tid * 128; off < bytes; off += 512u * 128u)
      __builtin_prefetch(p + off, 0, 1);
  }
  __syncthreads();

  // ---- layer 0: y = W0 · feats + b0 (K=3137: 3136 via WMMA + t column) ----
  {
    const int i0 = wave * 16;
    v8f c = wmma_matvec16(W0, kMlpIn, i0, sFeat, kMlpIn - 1);
    if (lane == 0 || lane == 16) {
      const int base = i0 + ((lane >> 4) << 3);
#pragma unroll
      for (int v = 0; v < 8; ++v) {
        const int r = base + v;
        sY[r] = c[v] + b0[r] + W0[(size_t)r * kMlpIn + (kMlpIn - 1)] * tval;
      }
    }
  }
  __syncthreads();
  rmsnorm_gelu(sY, sH, g, sRed, tid);

  // ---- hidden layers (256x256), fp32 WMMA ----
  for (int l = 0; l < 3; ++l) {
    const int i0 = wave * 16;
    v8f c = wmma_matvec16(Wh + (size_t)l * kHID * kHID, kHID, i0, sH, kHID);
    if (lane == 0 || lane == 16) {
      const int base = i0 + ((lane >> 4) << 3);
#pragma unroll
      for (int v = 0; v < 8; ++v) sY[base + v] = c[v] + bh[l * kHID + base + v];
    }
    __syncthreads();
    rmsnorm_gelu(sY, sH, g + (l + 1) * kHID, sRed, tid);
  }

  // ---- output layer: 3072x256, 192 tiles, 12 per wave, straight to d_out --
  for (int tb = wave; tb < kDim / 16; tb += 16) {
    const int i0 = tb * 16;
    v8f c = wmma_matvec16(Wout, kHID, i0, sH, kHID);
    if (lane == 0 || lane == 16) {
      const int base = i0 + ((lane >> 4) << 3);
#pragma unroll
      for (int v = 0; v < 8; ++v) out[base + v] = c[v] + bout[base + v];
    }
  }
}

} // anonymous namespace

extern "C" void kernel_launch(void* const* d_in, const int* in_sizes, int n_in,
                              void* d_out, int out_size, void* d_ws, size_t ws_size,
                              hipStream_t stream) {
  const float* x    = (const float*)d_in[0];
  const float* t    = (const float*)d_in[1];
  const float* Wk   = (const float*)d_in[2];
  // d_in[3] = bk: cancels inside softmax, unused.
  const float* Wv   = (const float*)d_in[4];
  const float* bv   = (const float*)d_in[5];
  const float* Wq   = (const float*)d_in[6];
  const float* bq   = (const float*)d_in[7];
  const float* W0   = (const float*)d_in[8];
  const float* b0   = (const float*)d_in[9];
  const float* Wh   = (const float*)d_in[10];
  const float* bh   = (const float*)d_in[11];
  const float* Wout = (const float*)d_in[12];
  const float* bout = (const float*)d_in[13];
  const float* g    = (const float*)d_in[14];

  float*    wsf = (float*)d_ws;
  unsigned* wsu = (unsigned*)d_ws;

  k_init<<<1, 256, 0, stream>>>(Wk, t, Wq, bq, wsf, wsu);
  k_dist<<<512, 256, 0, stream>>>(x, wsu);
  k_exps<<<512, 256, 0, stream>>>(x, wsf);
  k_mlp<<<1, 512, 0, stream>>>(x, t, Wv, bv, W0, b0, Wh, bh, Wout, bout, g,
                               wsf, (float*)d_out);
}